// SphericalHarmonics_47459388621149
// MI455X (gfx1250) — compile-verified
//
#include <hip/hip_runtime.h>
#include <math.h>

#define AS_GLOBAL __attribute__((address_space(1)))
#define AS_LDS    __attribute__((address_space(3)))

typedef float v4f __attribute__((ext_vector_type(4)));
typedef float v2f __attribute__((ext_vector_type(2)));
typedef int   v4i __attribute__((ext_vector_type(4)));

namespace {

constexpr int kNvr = 256;       // radial grid points (reference NVR)
constexpr int kNlm = 6;         // (l,m) pairs, l<=2
constexpr int kRowStride = 8;   // pad 6 -> 8 floats: 32B rows, b128+b64 friendly

// Per-pixel evaluation: cos(theta), 6 SH weights, uniform-grid interp of the
// 6 radial tables via two LDS row fetches, dual dot products.
__device__ __forceinline__ float sh_eval(float thv, float rv, const float* rows,
                                         float vr0, float inv_dvr,
                                         float vr_last, float tmax) {
  constexpr float kInv2Pi = 0.15915494309189535f;   // v_cos_f32 takes revolutions
  float x  = __builtin_amdgcn_cosf(thv * kInv2Pi);  // cos(theta)
  float s2 = fmaxf(1.0f - x * x, 0.0f);
  float s  = __builtin_amdgcn_sqrtf(s2);            // |sin(theta)| >= 0

  // Real Y_lm(theta=0 azimuth) with Condon-Shortley phase:
  float w0 =  0.28209479177387814f;                       // (0,0)
  float w1 =  0.4886025119029199f  * x;                   // (1,0)
  float w2 = -0.3454941494713355f  * s;                   // (1,1)
  float w3 =  0.31539156525252005f * fmaf(3.0f * x, x, -1.0f); // (2,0)
  float w4 = -0.7725484040463791f  * (x * s);             // (2,1)
  float w5 =  0.3862742020231896f  * s2;                  // (2,2)

  // Uniform-grid interpolation index (vr is a linspace).
  float t  = (rv - vr0) * inv_dvr;
  t        = fminf(fmaxf(t, 0.0f), tmax);       // left clamp -> fp[0]
  float fi = fminf(floorf(t), tmax - 1.0f);
  int   i  = (int)fi;
  float fr = t - fi;

  const float* rp = rows + i * kRowStride;
  v4f a4 = *(const v4f*)(rp);                    // ds_load_b128 (row i, k=0..3)
  v2f a2 = *(const v2f*)(rp + 4);                // ds_load_b64  (row i, k=4..5)
  v4f b4 = *(const v4f*)(rp + kRowStride);       // ds_load_b128 (row i+1)
  v2f b2 = *(const v2f*)(rp + kRowStride + 4);   // ds_load_b64  (row i+1)

  float da = fmaf(a4.x, w0, fmaf(a4.y, w1, fmaf(a4.z, w2,
             fmaf(a4.w, w3, fmaf(a2.x, w4, a2.y * w5)))));
  float db = fmaf(b4.x, w0, fmaf(b4.y, w1, fmaf(b4.z, w2,
             fmaf(b4.w, w3, fmaf(b2.x, w4, b2.y * w5)))));
  float res = fmaf(fr, db - da, da);

  // jnp.interp(..., right=1e-16): every radial table returns 1e-16 past vr[-1]
  float oor = 1e-16f * (w0 + w1 + w2 + w3 + w4 + w5);
  return (rv > vr_last) ? oor : res;
}

}  // namespace

__global__ __launch_bounds__(256) void SphericalHarmonics_47459388621149_kernel(
    const float* __restrict__ flm, const float* __restrict__ vr,
    const float* __restrict__ th, const float* __restrict__ rg,
    float* __restrict__ out, int n, int nvr) {
  __shared__ float raw[kNlm * kNvr];            // flm staged as-is (6 KB)
  __shared__ float rows[kNvr * kRowStride];     // transposed, padded rows (8 KB)
  const int tid = (int)threadIdx.x;

  // ---- Stage flm table into LDS via CDNA5 async global->LDS DMA ----
#if __has_builtin(__builtin_amdgcn_global_load_async_to_lds_b128)
  {
    const int xfers = (kNlm * kNvr) / 4;        // 384 x 16B transfers
    for (int tx = tid; tx < xfers; tx += (int)blockDim.x) {
      __builtin_amdgcn_global_load_async_to_lds_b128(
          (AS_GLOBAL v4i*)(const_cast<float*>(flm) + tx * 4),
          (AS_LDS v4i*)(&raw[tx * 4]), 0, 0);
    }
#if __has_builtin(__builtin_amdgcn_s_wait_asynccnt)
    __builtin_amdgcn_s_wait_asynccnt(0);
#else
    asm volatile("s_wait_asynccnt 0" ::: "memory");
#endif
  }
#else
  for (int j = tid; j < kNlm * kNvr; j += (int)blockDim.x) raw[j] = flm[j];
#endif
  __syncthreads();

  // Transpose into 32B-stride rows: rows[r][k] = flm[k][r]
  for (int r = tid; r < kNvr; r += (int)blockDim.x) {
#pragma unroll
    for (int k = 0; k < kNlm; ++k) rows[r * kRowStride + k] = raw[k * kNvr + r];
  }
  __syncthreads();

  const float vr0     = vr[0];
  const float inv_dvr = 1.0f / (vr[1] - vr[0]);
  const float vr_last = vr[nvr - 1];
  const float tmax    = (float)(nvr - 1);

  const int gsz  = (int)(gridDim.x * blockDim.x);
  const int gid  = (int)(blockIdx.x * blockDim.x) + tid;
  const int nvec = n >> 2;

  const v4f* th4 = (const v4f*)th;
  const v4f* rg4 = (const v4f*)rg;
  v4f*       ov4 = (v4f*)out;

  // Streaming main loop: 2x global_load_b128 + 1x global_store_b128 per 4 px.
  for (int v = gid; v < nvec; v += gsz) {
    __builtin_prefetch(&th4[v + gsz], 0, 0);    // global_prefetch_b8 (speculative)
    __builtin_prefetch(&rg4[v + gsz], 0, 0);
    v4f tv = th4[v];
    v4f rv = rg4[v];
    v4f o;
    o.x = sh_eval(tv.x, rv.x, rows, vr0, inv_dvr, vr_last, tmax);
    o.y = sh_eval(tv.y, rv.y, rows, vr0, inv_dvr, vr_last, tmax);
    o.z = sh_eval(tv.z, rv.z, rows, vr0, inv_dvr, vr_last, tmax);
    o.w = sh_eval(tv.w, rv.w, rows, vr0, inv_dvr, vr_last, tmax);
    ov4[v] = o;
  }
  // Scalar tail (n % 4 == 0 for the reference shapes, but stay generic).
  for (int p = (nvec << 2) + gid; p < n; p += gsz) {
    out[p] = sh_eval(th[p], rg[p], rows, vr0, inv_dvr, vr_last, tmax);
  }
}

extern "C" void kernel_launch(void* const* d_in, const int* in_sizes, int n_in,
                              void* d_out, int out_size, void* d_ws, size_t ws_size,
                              hipStream_t stream) {
  const float* flm = (const float*)d_in[0];   // [6, 256]
  const float* vr  = (const float*)d_in[1];   // [256]
  const float* th  = (const float*)d_in[2];   // [4096, 4096]
  const float* rg  = (const float*)d_in[3];   // [4096, 4096]
  float* out = (float*)d_out;                 // [4096, 4096]

  const int n   = in_sizes[2];
  const int nvr = in_sizes[1];

  const int threads = 256;
  int nvec   = n >> 2;
  int blocks = (nvec + threads - 1) / threads;
  if (blocks < 1) blocks = 1;
  if (blocks > 32768) blocks = 32768;   // grid-stride covers any remainder

  SphericalHarmonics_47459388621149_kernel<<<blocks, threads, 0, stream>>>(
      flm, vr, th, rg, out, n, nvr);
}